// TextureCreater_54924041782036
// MI455X (gfx1250) — compile-verified
//
#include <hip/hip_runtime.h>
#include <cstdint>

#define TS 1024
#define SS (TS * TS)
static constexpr float EPS = 1e-8f;

// CDNA5 async global->LDS path (gfx1250). Guarded so either toolchain compiles.
#if defined(__AMDGCN__) && __has_builtin(__builtin_amdgcn_global_load_async_to_lds_b128)
#define USE_ASYNC_LDS 1
#else
#define USE_ASYNC_LDS 0
#endif

#if USE_ASYNC_LDS
// Builtin expects GCC-vector int4 pointers (per hipcc diagnostic):
//   param0: 'int __attribute__((vector_size(16))) __device__ *'  (global src)
//   param1: LDS dst
typedef int v4i_gcc __attribute__((vector_size(16)));
typedef __attribute__((address_space(1))) v4i_gcc* as1_v4i;
typedef __attribute__((address_space(3))) v4i_gcc* as3_v4i;
#endif

static __device__ __forceinline__ int clampi(int v, int lo, int hi) {
  return v < lo ? lo : (v > hi ? hi : v);
}

// ---------------------------------------------------------------- zero
__global__ void zero4_kernel(float4* __restrict__ p, int total) {
  int i = blockIdx.x * blockDim.x + threadIdx.x;
  if (i < total) p[i] = make_float4(0.f, 0.f, 0.f, 0.f);
}

// ---------------------------------------------------------------- scatter splat
// acc[texel] = {wsum, r_acc, g_acc, b_acc}; 64MB for N=4 -> L2 resident.
__global__ void scatter_kernel(const float* __restrict__ img,
                               const float* __restrict__ uv,
                               float4* __restrict__ acc, int total) {
  int idx = blockIdx.x * blockDim.x + threadIdx.x;
  if (idx >= total) return;
  int n = idx >> 20;            // SS == 1<<20
  int p = idx & (SS - 1);
  const float sf = (float)(TS - 1);
  float u = uv[2 * idx] * sf;
  float v = sf - uv[2 * idx + 1] * sf;
  float x0f = floorf(u), y0f = floorf(v);
  float fx = u - x0f, fy = v - y0f;
  int x0 = (int)x0f, y0 = (int)y0f;
  const float* base = img + (size_t)n * 3 * SS + p;
  float r = base[0], g = base[SS], b = base[2 * SS];
  float wx[2] = {1.f - fx, fx};
  float wy[2] = {1.f - fy, fy};
  int xs[2] = {x0, x0 + 1};
  int ys[2] = {y0, y0 + 1};
  float4* accn = acc + (size_t)n * SS;
#pragma unroll
  for (int j = 0; j < 2; ++j)
#pragma unroll
    for (int i = 0; i < 2; ++i) {
      int xi = xs[i], yi = ys[j];
      float w = wx[i] * wy[j];
      // invalid corners get weight 0 in the reference -> skipping is identical
      if (xi >= 0 && xi < TS && yi >= 0 && yi < TS && w != 0.f) {
        float* t = (float*)(accn + ((size_t)yi * TS + xi));
        unsafeAtomicAdd(t + 0, w);
        unsafeAtomicAdd(t + 1, w * r);
        unsafeAtomicAdd(t + 2, w * g);
        unsafeAtomicAdd(t + 3, w * b);
      }
    }
}

// ---------------------------------------------------------------- resolve
// in: {w, r, g, b}  ->  out: {r/max(w,eps), g/.., b/.., mask}
__global__ void resolve_kernel(const float4* __restrict__ acc,
                               float4* __restrict__ tex, int total) {
  int i = blockIdx.x * blockDim.x + threadIdx.x;
  if (i >= total) return;
  float4 a = acc[i];
  float inv = 1.f / fmaxf(a.x, EPS);
  tex[i] = make_float4(a.y * inv, a.z * inv, a.w * inv, (a.x > 0.f) ? 1.f : 0.f);
}

// ---------------------------------------------------------------- 3x3 hole fill
// 32x8 block, 34x10 float4 halo tile staged in LDS (async-to-LDS on CDNA5).
template <bool FINAL>
__global__ void fill_kernel(const float4* __restrict__ in,
                            float4* __restrict__ out4,
                            float* __restrict__ out3) {
  __shared__ float4 tile[10][35];  // [row][col], col padded 34->35
  const int n = blockIdx.z;
  const int bx = blockIdx.x * 32;
  const int by = blockIdx.y * 8;
  const int tid = threadIdx.y * 32 + threadIdx.x;
  const float4* src = in + (size_t)n * SS;

  for (int e = tid; e < 340; e += 256) {  // 10 rows * 34 cols
    int tr = e / 34, tc = e % 34;
    int gy0 = by - 1 + tr, gx0 = bx - 1 + tc;
    int cy = clampi(gy0, 0, TS - 1);
    int cx = clampi(gx0, 0, TS - 1);
    const float4* gp = src + (size_t)cy * TS + cx;
#if USE_ASYNC_LDS
    __builtin_amdgcn_global_load_async_to_lds_b128(
        (as1_v4i)(uintptr_t)gp,
        (as3_v4i)&tile[tr][tc],
        /*offset=*/0, /*cpol=*/0);
#else
    tile[tr][tc] = *gp;
#endif
  }
#if USE_ASYNC_LDS
#if __has_builtin(__builtin_amdgcn_s_wait_asynccnt)
  __builtin_amdgcn_s_wait_asynccnt(0);
#else
  asm volatile("s_wait_asynccnt 0" ::: "memory");
#endif
#endif
  __syncthreads();

  const int lx = threadIdx.x + 1, ly = threadIdx.y + 1;
  const int gx = bx + threadIdx.x, gy = by + threadIdx.y;
  float4 c = tile[ly][lx];
  float nsr = 0.f, nsg = 0.f, nsb = 0.f, ncnt = 0.f;
#pragma unroll
  for (int dy = -1; dy <= 1; ++dy)
#pragma unroll
    for (int dx = -1; dx <= 1; ++dx) {
      int yy = gy + dy, xx = gx + dx;
      float4 t = tile[ly + dy][lx + dx];
      // zero-padding semantics: OOB neighbor contributes tex*m=0 and m=0
      float m = (yy >= 0 && yy < TS && xx >= 0 && xx < TS) ? t.w : 0.f;
      nsr += t.x * m;
      nsg += t.y * m;
      nsb += t.z * m;
      ncnt += m;
    }
  float inv = 1.f / fmaxf(ncnt, EPS);
  bool newmask = ncnt > 0.f;
  bool mask = c.w > 0.f;
  float fr = newmask ? nsr * inv : 0.f;
  float fg = newmask ? nsg * inv : 0.f;
  float fb = newmask ? nsb * inv : 0.f;
  float orr = mask ? c.x : fr;
  float og = mask ? c.y : fg;
  float ob = mask ? c.z : fb;
  if (FINAL) {
    size_t o = (((size_t)n * TS + gy) * TS + gx) * 3;
    out3[o + 0] = orr;
    out3[o + 1] = og;
    out3[o + 2] = ob;
  } else {
    out4[(size_t)n * SS + (size_t)gy * TS + gx] =
        make_float4(orr, og, ob, (mask || newmask) ? 1.f : 0.f);
  }
}

// ---------------------------------------------------------------- launch
extern "C" void kernel_launch(void* const* d_in, const int* in_sizes, int n_in,
                              void* d_out, int out_size, void* d_ws, size_t ws_size,
                              hipStream_t stream) {
  (void)n_in; (void)out_size; (void)ws_size;
  const float* img = (const float*)d_in[0];  // [N,3,1024,1024] f32
  const float* uv  = (const float*)d_in[1];  // [N,1024,1024,2] f32
  float* out = (float*)d_out;                // [N,1024,1024,3] f32

  const int N = in_sizes[0] / (3 * SS);
  const int total = N * SS;

  float4* bufA = (float4*)d_ws;              // acc, then fill ping
  float4* bufB = bufA + (size_t)total;       // resolved tex, fill pong

  const int threads = 256;
  const int blocks = (total + threads - 1) / threads;

  zero4_kernel<<<blocks, threads, 0, stream>>>(bufA, total);
  scatter_kernel<<<blocks, threads, 0, stream>>>(img, uv, bufA, total);
  resolve_kernel<<<blocks, threads, 0, stream>>>(bufA, bufB, total);

  dim3 g(TS / 32, TS / 8, N), b(32, 8);
  fill_kernel<false><<<g, b, 0, stream>>>(bufB, bufA, nullptr);
  fill_kernel<true><<<g, b, 0, stream>>>(bufA, nullptr, out);
}